// Net4_57767310131827
// MI455X (gfx1250) — compile-verified
//
#include <hip/hip_runtime.h>
#include <math.h>

typedef __attribute__((ext_vector_type(2))) float v2f;
typedef __attribute__((ext_vector_type(8))) float v8f;

#define LOG_2PI 1.8378770664093453f
#define ACT_ELEMS 6291456   // 512*12288, constant across all layer tensors
#define PER_BATCH 12288

// ---------------------------------------------------------------------------
__global__ void k_init(float* lsum) {
  if (threadIdx.x < 4) lsum[threadIdx.x] = 0.0f;
}

// ---------------------------------------------------------------------------
// logit (with logdet) fused with first squeeze:
// x (512,3,64,64) -> out (512,12,32,32); bsum[b] = sum(-log(xs)-log(1-xs))
__global__ void k_logit_squeeze(const float* __restrict__ x,
                                float* __restrict__ out,
                                float* __restrict__ bsum) {
  const int b = blockIdx.x;
  const float* xb = x + (size_t)b * PER_BATCH;
  float* ob = out + (size_t)b * PER_BATCH;
  float acc = 0.0f;
  for (int idx = threadIdx.x; idx < PER_BATCH; idx += blockDim.x) {
    int co = idx >> 10;           // out channel 0..11  (32*32=1024 pixels)
    int rest = idx & 1023;
    int h = rest >> 5, w = rest & 31;
    int c = co >> 2, s0 = (co >> 1) & 1, s1 = co & 1;
    float xv = xb[(c * 64 + (2 * h + s0)) * 64 + (2 * w + s1)];
    float xs = 0.0005f + xv * 0.999f;
    float la = logf(xs), lb = logf(1.0f - xs);
    ob[idx] = la - lb;
    acc += -(la + lb);
  }
  __shared__ float red[256];
  red[threadIdx.x] = acc;
  __syncthreads();
  for (int s = blockDim.x >> 1; s > 0; s >>= 1) {
    if ((int)threadIdx.x < s) red[threadIdx.x] += red[threadIdx.x + s];
    __syncthreads();
  }
  if (threadIdx.x == 0) bsum[b] = red[0];
}

// ---------------------------------------------------------------------------
// leaky-relu (with logdet, replicating reference formula) fused with squeeze:
// in (512,C,n,n) -> out (512,4C,n/2,n/2); bsum[b] += sum(log(deriv))
template <int C, int N>
__global__ void k_lrelu_squeeze(const float* __restrict__ in,
                                float* __restrict__ out,
                                float* __restrict__ bsum) {
  constexpr int N2 = N / 2;
  constexpr int HW2 = N2 * N2;          // power of two
  const int b = blockIdx.x;
  const float* ib = in + (size_t)b * PER_BATCH;
  float* ob = out + (size_t)b * PER_BATCH;
  float acc = 0.0f;
  for (int idx = threadIdx.x; idx < PER_BATCH; idx += blockDim.x) {
    int co = idx / HW2;                 // constant divisor -> shift
    int rest = idx & (HW2 - 1);
    int h = rest / N2, w = rest & (N2 - 1);
    int c = co >> 2, s0 = (co >> 1) & 1, s1 = co & 1;
    float v = ib[((size_t)c * N + (2 * h + s0)) * N + (2 * w + s1)];
    float vp = fmaxf(v, 0.0f);
    float r = vp / (v + 0.001f);           // x_ge_zero exactly as reference
    float d1 = 1.2f * r;
    float deriv = d1 + 0.8f * (1.0f - d1); // reference's exact expression
    ob[idx] = 1.2f * vp + 0.8f * (v - vp);
    acc += logf(deriv);
  }
  __shared__ float red[256];
  red[threadIdx.x] = acc;
  __syncthreads();
  for (int s = blockDim.x >> 1; s > 0; s >>= 1) {
    if ((int)threadIdx.x < s) red[threadIdx.x] += red[threadIdx.x + s];
    __syncthreads();
  }
  if (threadIdx.x == 0) bsum[b] += red[0];
}

// ---------------------------------------------------------------------------
// Circular 3x3 conv as implicit GEMM on the fp32 WMMA pipe.
// GEMM: D[M=512*N*N][C] = A[M][K=9C] * W[K][C] + bias, K ordered k = t*C + ci
// (t = kernel tap 0..8). C%4==0 so a K-chunk of 4 never crosses a tap:
// with the tap loop unrolled, all spatial wrap math is loop-invariant and the
// inner channel loop is pure strided addressing (no integer division; weight
// loads for the 9 taps of one channel are contiguous -> b128 clauses).
// Each wave register-blocks MT x NT 16x16 tiles -> MT*NT independent
// accumulator chains hide V_WMMA_F32_16X16X4_F32 pipe latency.
// Fragment layout (ISA 7.12.2, 32-bit):
//   A 16x4 : lanes0-15 K=k0,k0+1 (vgpr0,1); lanes16-31 K=k0+2,k0+3
//   B 4x16 : same K split, N = lane&15
//   C/D    : vgpr v -> row v / v+8 per lane half, col = lane&15
template <int C, int N, int MT, int NT>
__global__ void k_conv_wmma(const float* __restrict__ in,    // (512,C,N,N)
                            const float* __restrict__ kern,  // (C,C,3,3)
                            const float* __restrict__ bias,  // (C)
                            float* __restrict__ out) {       // (512,C,N,N)
  constexpr int HW = N * N;                 // power of two
  constexpr int M = 512 * HW;
  constexpr int Mtiles = M / 16;
  constexpr int Ntiles = (C + 15) / 16;
  constexpr int MG = Mtiles / MT;
  constexpr int NG = (Ntiles + NT - 1) / NT;
  constexpr int K9 = 9 * C;
  constexpr bool FULLN = (C % 16 == 0);

  const int lane = threadIdx.x & 31;
  const int half = lane >> 4;
  const int lm = lane & 15;
  const int waveId = blockIdx.x * (blockDim.x >> 5) + (threadIdx.x >> 5);
  const int nWaves = gridDim.x * (blockDim.x >> 5);

  for (int g = waveId; g < MG * NG; g += nWaves) {
    const int ng = g % NG;
    const int mg = g / NG;

    int hA[MT], wA[MT];
    const float* inb[MT];
#pragma unroll
    for (int i = 0; i < MT; ++i) {
      const int rowA = (mg * MT + i) * 16 + lm;
      const int bA = rowA / HW;             // shift
      const int pix = rowA & (HW - 1);
      hA[i] = pix / N;                      // shift
      wA[i] = pix & (N - 1);
      inb[i] = in + (size_t)bA * C * HW;
    }

    int ncol[NT];
    bool nvalid[NT];
    const float* kcol[NT];
#pragma unroll
    for (int j = 0; j < NT; ++j) {
      ncol[j] = (ng * NT + j) * 16 + lm;
      nvalid[j] = FULLN ? true : (ncol[j] < C);
      kcol[j] = kern + (size_t)(nvalid[j] ? ncol[j] : 0) * K9;
    }

    v8f acc[MT][NT];
#pragma unroll
    for (int i = 0; i < MT; ++i)
#pragma unroll
      for (int j = 0; j < NT; ++j)
        acc[i][j] = (v8f){};

#pragma unroll
    for (int t = 0; t < 9; ++t) {
      const int kh = t / 3, kw = t % 3;     // constants after unroll
      int base[MT];
#pragma unroll
      for (int i = 0; i < MT; ++i) {
        const int hh = (hA[i] + kh - 1 + N) & (N - 1);  // circular wrap
        const int ww = (wA[i] + kw - 1 + N) & (N - 1);
        base[i] = hh * N + ww;
      }
      for (int c0 = 0; c0 < C; c0 += 4) {
        const int kA = c0 + half * 2;       // this lane-half's channel pair
        v2f a[MT], bf[NT];
#pragma unroll
        for (int i = 0; i < MT; ++i) {
          a[i][0] = inb[i][(size_t)(kA + 0) * HW + base[i]];
          a[i][1] = inb[i][(size_t)(kA + 1) * HW + base[i]];
        }
#pragma unroll
        for (int j = 0; j < NT; ++j) {
          const float b0 = kcol[j][(kA + 0) * 9 + t];
          const float b1 = kcol[j][(kA + 1) * 9 + t];
          bf[j][0] = nvalid[j] ? b0 : 0.0f;
          bf[j][1] = nvalid[j] ? b1 : 0.0f;
        }
#pragma unroll
        for (int i = 0; i < MT; ++i)
#pragma unroll
          for (int j = 0; j < NT; ++j)
            acc[i][j] = __builtin_amdgcn_wmma_f32_16x16x4_f32(
                false, a[i], false, bf[j], (short)0, acc[i][j], false, false);
      }
    }

#pragma unroll
    for (int j = 0; j < NT; ++j) {
      if (!nvalid[j]) continue;
      const float bv = bias[ncol[j]];
#pragma unroll
      for (int i = 0; i < MT; ++i) {
#pragma unroll
        for (int v = 0; v < 8; ++v) {
          const int r = (mg * MT + i) * 16 + v + 8 * half;
          const int bD = r / HW;            // shift
          const int pix = r & (HW - 1);
          out[((size_t)bD * C + ncol[j]) * HW + pix] = acc[i][j][v] + bv;
        }
      }
    }
  }
}

// ---------------------------------------------------------------------------
// Khat[u,v][co][ci] = sum_{kh,kw} K[co,ci,kh,kw] * exp(-2*pi*i*(u*kh+v*kw)/n)
__global__ void k_khat(const float* __restrict__ kern, float* __restrict__ KH,
                       int C, int n) {
  const long total = (long)n * n * C * C;
  const long id = (long)blockIdx.x * blockDim.x + threadIdx.x;
  if (id >= total) return;
  const int ci = (int)(id % C);
  const long t1 = id / C;
  const int co = (int)(t1 % C);
  const long uv = t1 / C;
  const int v = (int)(uv % n);
  const int u = (int)(uv / n);
  const float* kk = kern + ((size_t)co * C + ci) * 9;
  const float w0 = -6.283185307179586f / (float)n;
  float re = 0.0f, im = 0.0f;
  for (int kh = 0; kh < 3; ++kh)
    for (int kw = 0; kw < 3; ++kw) {
      float s, c;
      sincosf(w0 * (float)(u * kh + v * kw), &s, &c);
      const float kv = kk[kh * 3 + kw];
      re += kv * c;
      im += kv * s;
    }
  KH[2 * id] = re;
  KH[2 * id + 1] = im;
}

// ---------------------------------------------------------------------------
// slogdet via complex LU (no pivoting; Khat ~ identity DFT + 1% noise).
// One workgroup per frequency.
__global__ void k_logdet_lu(float* __restrict__ KH, float* __restrict__ lsum,
                            int C, int layer) {
  float* A = KH + (size_t)blockIdx.x * C * C * 2;
  __shared__ float piv[2];
  __shared__ float lds_ld;
  if (threadIdx.x == 0) lds_ld = 0.0f;
  __syncthreads();
  for (int j = 0; j < C; ++j) {
    if (threadIdx.x == 0) {
      const float dre = A[2 * (j * C + j)], dim = A[2 * (j * C + j) + 1];
      const float mag2 = dre * dre + dim * dim;
      lds_ld += 0.5f * logf(mag2);
      const float inv = 1.0f / mag2;
      piv[0] = dre * inv;
      piv[1] = -dim * inv;
    }
    __syncthreads();
    const float ire = piv[0], iim = piv[1];
    for (int i = j + 1 + threadIdx.x; i < C; i += blockDim.x) {
      const float are = A[2 * (i * C + j)], aim = A[2 * (i * C + j) + 1];
      const float fre = are * ire - aim * iim;
      const float fim = are * iim + aim * ire;
      for (int k = j + 1; k < C; ++k) {
        const float ure = A[2 * (j * C + k)], uim = A[2 * (j * C + k) + 1];
        A[2 * (i * C + k)]     -= fre * ure - fim * uim;
        A[2 * (i * C + k) + 1] -= fre * uim + fim * ure;
      }
    }
    __syncthreads();
  }
  if (threadIdx.x == 0) atomicAdd(&lsum[layer], lds_ld);
}

// ---------------------------------------------------------------------------
// log_pdf_x[b] = sum(-0.5*y^2 - 0.5*log(2pi)) + bsum[b] + lsum[0..2]
__global__ void k_final(const float* __restrict__ y,
                        const float* __restrict__ bsum,
                        const float* __restrict__ lsum,
                        float* __restrict__ lp) {
  const int b = blockIdx.x;
  const float* yb = y + (size_t)b * PER_BATCH;
  float acc = 0.0f;
  for (int i = threadIdx.x; i < PER_BATCH; i += blockDim.x) {
    const float v = yb[i];
    acc += -0.5f * v * v;
  }
  __shared__ float red[256];
  red[threadIdx.x] = acc;
  __syncthreads();
  for (int s = blockDim.x >> 1; s > 0; s >>= 1) {
    if ((int)threadIdx.x < s) red[threadIdx.x] += red[threadIdx.x + s];
    __syncthreads();
  }
  if (threadIdx.x == 0)
    lp[b] = red[0] - 0.5f * LOG_2PI * (float)PER_BATCH + bsum[b] +
            lsum[0] + lsum[1] + lsum[2];
}

// ---------------------------------------------------------------------------
extern "C" void kernel_launch(void* const* d_in, const int* in_sizes, int n_in,
                              void* d_out, int out_size, void* d_ws, size_t ws_size,
                              hipStream_t stream) {
  const float* x  = (const float*)d_in[0];
  const float* k1 = (const float*)d_in[1];
  const float* b1 = (const float*)d_in[2];
  const float* k2 = (const float*)d_in[3];
  const float* b2 = (const float*)d_in[4];
  const float* k3 = (const float*)d_in[5];
  const float* b3 = (const float*)d_in[6];

  float* y  = (float*)d_out;             // 6,291,456 floats
  float* lp = (float*)d_out + ACT_ELEMS; // 512 floats

  float* ws    = (float*)d_ws;
  float* actA  = ws;                       // 6.29M floats
  float* actB  = ws + ACT_ELEMS;           // 6.29M floats
  float* KH    = ws + 2 * ACT_ELEMS;       // up to 64*192*192*2 = 4,718,592
  float* bsum  = KH + 4718592;             // 512
  float* lsum  = bsum + 512;               // 4

  k_init<<<1, 32, 0, stream>>>(lsum);

  // main chain (wave-groups: L1 16384, L2 4096, L3 6144; 8 waves/block)
  k_logit_squeeze<<<512, 256, 0, stream>>>(x, actA, bsum);
  k_conv_wmma<12, 32, 2, 1><<<2048, 256, 0, stream>>>(actA, k1, b1, actB);
  k_lrelu_squeeze<12, 32><<<512, 256, 0, stream>>>(actB, actA, bsum);
  k_conv_wmma<48, 16, 2, 3><<<512, 256, 0, stream>>>(actA, k2, b2, actB);
  k_lrelu_squeeze<48, 16><<<512, 256, 0, stream>>>(actB, actA, bsum);
  k_conv_wmma<192, 8, 2, 2><<<768, 256, 0, stream>>>(actA, k3, b3, y);

  // conv operator logdets (batch-independent scalars)
  k_khat<<<(1024 * 144 + 255) / 256, 256, 0, stream>>>(k1, KH, 12, 32);
  k_logdet_lu<<<1024, 64, 0, stream>>>(KH, lsum, 12, 0);
  k_khat<<<(256 * 2304 + 255) / 256, 256, 0, stream>>>(k2, KH, 48, 16);
  k_logdet_lu<<<256, 128, 0, stream>>>(KH, lsum, 48, 1);
  k_khat<<<(64 * 36864 + 255) / 256, 256, 0, stream>>>(k3, KH, 192, 8);
  k_logdet_lu<<<64, 256, 0, stream>>>(KH, lsum, 192, 2);

  k_final<<<512, 256, 0, stream>>>(y, bsum, lsum, lp);
}